// SignedDistanceTransform_74981539053936
// MI455X (gfx1250) — compile-verified
//
#include <hip/hip_runtime.h>
#include <math.h>

// ---------------------------------------------------------------------------
// Signed distance transform, 192^3, exact separable min-plus squared EDT.
//   d[i] = min_j f[j] + (i-j)^2
// One V_WMMA_F32_16X16X4_F32 produces the full 16x16 tile directly:
//   A[i][:] = [1, i, i^2, 0] ,  B[:][j] = [f[j]+j^2, -2j, 1, 0]^T , C = 0
//   => D[i][j] = f[j] + j^2 - 2ij + i^2 = f[j] + (i-j)^2       (exact in f32)
// Min over j = v_min3 chains across 12 j-tiles + ds_swizzle xor butterfly.
// Lines are staged global<->LDS with CDNA5 async-copy ops (ASYNCcnt).
// ---------------------------------------------------------------------------

typedef __attribute__((ext_vector_type(2))) float v2f;
typedef __attribute__((ext_vector_type(8))) float v8f;

#define GAS __attribute__((address_space(1)))
#define LAS __attribute__((address_space(3)))

#define NLIN  192
#define NVOX  (192 * 192 * 192)
#define BIGF  1.0e8f

__device__ __forceinline__ void async_load_f32(const float* g, float* l) {
  __builtin_amdgcn_global_load_async_to_lds_b32((GAS int*)g, (LAS int*)l, 0, 0);
}
__device__ __forceinline__ void async_store_f32(float* g, const float* l) {
  __builtin_amdgcn_global_store_async_from_lds_b32((GAS int*)g, (LAS int*)l, 0, 0);
}

// xor-butterfly min over the 16 lanes of each half-wave (wave32, group-of-32
// ds_swizzle: offset = xor_mask<<10 | and_mask(0x1f))
__device__ __forceinline__ float halfmin16(float v) {
  v = fminf(v, __int_as_float(__builtin_amdgcn_ds_swizzle(__float_as_int(v), 0x041f)));
  v = fminf(v, __int_as_float(__builtin_amdgcn_ds_swizzle(__float_as_int(v), 0x081f)));
  v = fminf(v, __int_as_float(__builtin_amdgcn_ds_swizzle(__float_as_int(v), 0x101f)));
  v = fminf(v, __int_as_float(__builtin_amdgcn_ds_swizzle(__float_as_int(v), 0x201f)));
  return v;
}

// One wave (32 lanes) computes the exact 1D min-plus transform of a 192-line
// held in LDS. 144 WMMAs: 12 i-tiles x 12 j-tiles.
__device__ __noinline__ void minplus_wave(const float* lin, float* lout, int lane) {
  const bool lo = (lane < 16);
  const int  jl = lane & 15;

  // B operands, one per j-tile (constant over i-tiles). f32 B 4x16 layout:
  // lanes 0-15 hold rows K=0,1 (vgpr0,1); lanes 16-31 hold rows K=2,3.
  //   row0 = f[j]+j^2 , row1 = -2j , row2 = 1 , row3 = 0
  v2f B[12];
#pragma unroll
  for (int jt = 0; jt < 12; ++jt) {
    int   j  = jt * 16 + jl;
    float fj = lin[j];                       // LDS read (hi half reads too; unused)
    B[jt][0] = lo ? (fj + (float)(j * j)) : 1.0f;
    B[jt][1] = lo ? (-2.0f * (float)j) : 0.0f;
  }

  const v8f cz = {0.f, 0.f, 0.f, 0.f, 0.f, 0.f, 0.f, 0.f};

#pragma unroll
  for (int it = 0; it < 12; ++it) {
    const int   ib = it * 16;
    const float fi = (float)(ib + jl);
    // A 16x4 layout: lanes 0-15 hold K=0,1 (vgpr0,1); lanes 16-31 hold K=2,3.
    //   A[i] = [1, i, i^2, 0]
    v2f A;
    A[0] = lo ? 1.0f : fi * fi;
    A[1] = lo ? fi : 0.0f;

    v8f m = {1e30f, 1e30f, 1e30f, 1e30f, 1e30f, 1e30f, 1e30f, 1e30f};
#pragma unroll
    for (int jt = 0; jt < 12; ++jt) {
      // D[i][j] = f[j] + (i-j)^2
      v8f d = __builtin_amdgcn_wmma_f32_16x16x4_f32(
          false, A, false, B[jt], (short)0, cz, false, false);
#pragma unroll
      for (int r = 0; r < 8; ++r) m[r] = fminf(m[r], d[r]);
    }

    // Row-min across the 16 lanes holding columns j=0..15 of each row.
    // Lanes 0-15 carry rows ib..ib+7, lanes 16-31 carry rows ib+8..ib+15.
#pragma unroll
    for (int r = 0; r < 8; ++r) {
      float v = halfmin16(m[r]);
      int   i = ib + r + (lo ? 0 : 8);
      lout[i] = v;                           // 16 lanes store same value/addr
    }
  }
}

// ---- seed init: fg = (x==0)?0:INF ; bg = (x!=0)?0:INF ----------------------
__global__ void __launch_bounds__(256) sdt_init(const float* __restrict__ x,
                                                float* __restrict__ fg,
                                                float* __restrict__ bg) {
  int i = blockIdx.x * 256 + threadIdx.x;
  float v = x[i];
  fg[i] = (v == 0.0f) ? 0.0f : BIGF;
  bg[i] = (v != 0.0f) ? 0.0f : BIGF;
}

// ---- pass along the contiguous axis (z): one wave per line -----------------
__global__ void __launch_bounds__(256) sdt_pass_contig(float* volFG, float* volBG) {
  __shared__ float bin[8][NLIN];
  __shared__ float bout[8][NLIN];
  const int w = threadIdx.x >> 5;
  const int t = threadIdx.x & 31;
  float* vol  = (blockIdx.y == 0) ? volFG : volBG;
  float* base = vol + (size_t)(blockIdx.x * 8 + w) * NLIN;

#pragma unroll
  for (int k = 0; k < 6; ++k) {             // lane t moves elements t+32k
    int e = t + 32 * k;
    async_load_f32(base + e, &bin[w][e]);
  }
  asm volatile("s_wait_asynccnt 0" ::: "memory");

  minplus_wave(&bin[w][0], &bout[w][0], t);

  asm volatile("s_wait_dscnt 0" ::: "memory");
#pragma unroll
  for (int k = 0; k < 6; ++k) {
    int e = t + 32 * k;
    async_store_f32(base + e, &bout[w][e]);
  }
  // S_ENDPGM performs implicit wait-idle, draining ASYNCcnt.
}

// ---- pass along a strided axis: 32 lines batched over contiguous z ---------
// element j of line t:  addr = u*uStride + j*elemStride + z0 + t  (coalesced).
__global__ void __launch_bounds__(256) sdt_pass_strided(float* volFG, float* volBG,
                                                        int elemStride, int uStride) {
  __shared__ float bin[32][NLIN + 1];       // 193-float pitch: conflict-free
  __shared__ float bout[32][NLIN + 1];
  const int w = threadIdx.x >> 5;
  const int t = threadIdx.x & 31;
  float* vol   = (blockIdx.y == 0) ? volFG : volBG;
  const int u  = blockIdx.x / 6;
  const int z0 = (blockIdx.x % 6) * 32;
  float* gbase = vol + (size_t)u * (size_t)uStride + (size_t)(z0 + t);

  for (int j = w; j < NLIN; j += 8) {       // block-cooperative coalesced load
    async_load_f32(gbase + (size_t)j * (size_t)elemStride, &bin[t][j]);
  }
  asm volatile("s_wait_asynccnt 0" ::: "memory");
  __syncthreads();

  for (int q = 0; q < 4; ++q) {             // each wave transforms 4 lines
    int ln = w * 4 + q;
    minplus_wave(&bin[ln][0], &bout[ln][0], t);
  }
  __syncthreads();

  for (int j = w; j < NLIN; j += 8) {       // block-cooperative coalesced store
    async_store_f32(gbase + (size_t)j * (size_t)elemStride, &bout[t][j]);
  }
}

// ---- final: tanh((sqrt(fg)-sqrt(bg))/SIGMA) --------------------------------
__global__ void __launch_bounds__(256) sdt_final(const float* __restrict__ fg,
                                                 const float* __restrict__ bg,
                                                 float* __restrict__ out) {
  int i = blockIdx.x * 256 + threadIdx.x;
  float a = sqrtf(fg[i]);
  float b = sqrtf(bg[i]);
  out[i]  = tanhf((a - b) * 0.1f);
}

extern "C" void kernel_launch(void* const* d_in, const int* in_sizes, int n_in,
                              void* d_out, int out_size, void* d_ws, size_t ws_size,
                              hipStream_t stream) {
  (void)in_sizes; (void)n_in; (void)out_size; (void)ws_size;
  const float* x  = (const float*)d_in[0];
  float*       fg = (float*)d_out;          // fg squared-distance volume
  float*       bg = (float*)d_ws;           // bg squared-distance volume (28.3 MB)

  dim3 blk(256);
  sdt_init<<<dim3(NVOX / 256), blk, 0, stream>>>(x, fg, bg);

  // axis 2 (z, contiguous): 36864 lines, 8 per block, both volumes via grid.y
  sdt_pass_contig<<<dim3(36864 / 8, 2), blk, 0, stream>>>(fg, bg);
  // axis 1 (y): elemStride=192,   u=x (stride 36864), batch over z
  sdt_pass_strided<<<dim3(192 * 6, 2), blk, 0, stream>>>(fg, bg, 192, 36864);
  // axis 0 (x): elemStride=36864, u=y (stride 192),   batch over z
  sdt_pass_strided<<<dim3(192 * 6, 2), blk, 0, stream>>>(fg, bg, 36864, 192);

  sdt_final<<<dim3(NVOX / 256), blk, 0, stream>>>(fg, bg, (float*)d_out);
}